// NoTradeRegionRNN_9242769622585
// MI455X (gfx1250) — compile-verified
//
#include <hip/hip_runtime.h>

// NoTradeRegionRNN on gfx1250 (MI455X).
// Persistent-tile design: 32 workgroups x 256 threads (8 wave32s).
// Each workgroup owns a 16-column batch tile for all 63 timesteps.
// v (128x128 f32) lives in LDS; each wave register-caches its 16-row
// A fragments (32 x v2f). 128x16 activation tiles are stored pair-row
// interleaved so every B fragment is ONE aligned ds_load_b64, feeding
// two interleaved V_WMMA_F32_16X16X4_F32 accumulator chains.
// pi-bar handling is folded into a precomputed bias: v@(x-pi) = v@x - bias.

#define DD   128
#define TT   64
#define BB   512
#define BT   16            // batch columns per workgroup
#define TB   (TT * BB)     // stride of d index in (D,T,B) tensors
#define VP   132           // v row stride (conflict-friendly, 16B aligned)
#define PRS  34            // pair-row stride (floats): 64 pair-rows x 34
#define NTE  (64 * PRS)    // tile element count
#define EPSF 1e-5f

// pair-row interleaved tile index: (row i, col c) -> flat float index
#define XI(i, c) ((((i) >> 1) * PRS) + ((c) << 1) + ((i) & 1))

typedef __attribute__((ext_vector_type(2))) float v2f;
typedef __attribute__((ext_vector_type(8))) float v8f;

// Raw product slice: wave w computes rows [16w,16w+16) of v @ X.
// A fragments pre-cached in registers; B-frag = one b64 from the
// pair-interleaved tile. Two accumulator chains for ILP.
__device__ __forceinline__ v8f wmma_vx(const v2f* __restrict__ aF,
                                       const float* __restrict__ xL,
                                       int half, int lr) {
  const v2f* xp = (const v2f*)xL;      // pair-row p at xp[p*17 + lr]
  v8f acc0 = {}, acc1 = {};
#pragma unroll
  for (int kk = 0; kk < 32; kk += 2) {
    v2f b0 = xp[(2 * kk + half) * (PRS / 2) + lr];
    v2f b1 = xp[(2 * kk + 2 + half) * (PRS / 2) + lr];
    acc0 = __builtin_amdgcn_wmma_f32_16x16x4_f32(false, aF[kk], false, b0,
                                                 (short)0, acc0, false, false);
    acc1 = __builtin_amdgcn_wmma_f32_16x16x4_f32(false, aF[kk + 1], false, b1,
                                                 (short)0, acc1, false, false);
  }
  return acc0 + acc1;
}

__global__ __launch_bounds__(256)
void notrade_rnn_kernel(const float* __restrict__ inp,    // (D,T,B)
                        const float* __restrict__ target, // (D)
                        const float* __restrict__ rets,   // (D,T,B)
                        const float* __restrict__ W,      // (D,D)
                        const float* __restrict__ bvec,   // (D)
                        float* __restrict__ out,          // (D,T,B)
                        float* __restrict__ hlast) {      // (D,B)
  __shared__ float vLds[DD * VP];
  __shared__ __attribute__((aligned(16))) float hCur[NTE];   // h -> adj -> h_next
  __shared__ __attribute__((aligned(16))) float htLds[NTE];  // Gauss-Seidel h_temp
  __shared__ __attribute__((aligned(16))) float sLds[NTE];   // s, then GS-updated s
  __shared__ __attribute__((aligned(16))) float dLds[NTE];   // delta for judge test
  __shared__ __attribute__((aligned(16))) float hinLds[NTE];
  __shared__ __attribute__((aligned(16))) float houtLds[NTE];
  __shared__ __attribute__((aligned(16))) float hmLds[NTE];
  __shared__ float absbLds[DD];
  __shared__ float targLds[DD];
  __shared__ float biasLds[DD];     // bias[i] = v[i,:] . target
  __shared__ float denomLds[BT];
  __shared__ float djLds[BT];
  __shared__ float judgeLds[BT];
  __shared__ int   jmaskLds[BT];
  __shared__ int   flagPar[2];
  __shared__ int   allJ;

  const int tid     = threadIdx.x;
  const int colBase = blockIdx.x * BT;
  const int col     = tid & 15;     // batch column within tile
  const int ig      = tid >> 4;     // 0..15, row group (8 rows each)
  const int gb      = colBase + col;
  const int wave    = tid >> 5;
  const int lane    = tid & 31;
  const int half    = lane >> 4;
  const int lr      = lane & 15;
  const int mb      = wave * 16 + half * 8;  // accumulator row base per lane
  const int mrow    = wave * 16 + lr;        // A-frag row per lane

  // ---- setup phase A: absb, pi_bar, v = W / max(rowNorm, 1e-12) ----
  if (tid < DD) {
    absbLds[tid] = fabsf(bvec[tid]);
    targLds[tid] = target[tid];
  }
  {
    const int row = tid >> 1;
    const int hs  = (tid & 1) * 64;
    float ss = 0.f;
    for (int j = hs; j < hs + 64; ++j) {
      float w = W[row * DD + j];
      ss += w * w;
    }
    ss += __shfl_xor(ss, 1, 32);
    const float inv = 1.0f / fmaxf(sqrtf(ss), 1e-12f);
    for (int j = hs; j < hs + 64; ++j)
      vLds[row * VP + j] = W[row * DD + j] * inv;
  }
  __syncthreads();

  // ---- setup phase B: bias = v @ pi_bar ; h0 load/emit ----
  {
    const int row = tid >> 1;
    const int hs  = (tid & 1) * 64;
    float sb = 0.f;
    for (int j = hs; j < hs + 64; ++j)
      sb += vLds[row * VP + j] * targLds[j];
    sb += __shfl_xor(sb, 1, 32);
    if ((tid & 1) == 0) biasLds[row] = sb;
  }
#pragma unroll
  for (int q = 0; q < 8; ++q) {
    const int i    = ig * 8 + q;
    const float h0 = inp[i * TB + gb];
    hCur[XI(i, col)] = h0;
    out[i * TB + gb] = h0;
  }
  __syncthreads();

  // ---- per-wave register caches: A fragments, bias, absb ----
  v2f aFrag[32];
#pragma unroll
  for (int kk = 0; kk < 32; ++kk) {
    aFrag[kk].x = vLds[mrow * VP + kk * 4 + 2 * half];
    aFrag[kk].y = vLds[mrow * VP + kk * 4 + 2 * half + 1];
  }
  float biasReg[8], absbReg[8];
#pragma unroll
  for (int r = 0; r < 8; ++r) {
    biasReg[r] = biasLds[mb + r];
    absbReg[r] = absbLds[mb + r];
  }

  // ---- sequential scan over T-1 steps ----
  for (int t = 0; t < TT - 1; ++t) {
    // (1) adj = h*(1+r)/(1+sum(h*r)); prefetch next step's r slice
    float rv[8];
    float partial = 0.f;
#pragma unroll
    for (int q = 0; q < 8; ++q) {
      const int i = ig * 8 + q;
      rv[q] = rets[i * TB + t * BB + gb];
      partial += hCur[XI(i, col)] * rv[q];
    }
    if (t + 1 < TT - 1) {
#pragma unroll
      for (int q = 0; q < 8; ++q)
        __builtin_prefetch(&rets[(ig * 8 + q) * TB + (t + 1) * BB + gb], 0, 1);
    }
    if (tid < BT) denomLds[tid] = 1.0f;
    __syncthreads();
    atomicAdd(&denomLds[col], partial);
    __syncthreads();
    const float dn = denomLds[col];
#pragma unroll
    for (int q = 0; q < 8; ++q) {
      const int i = ig * 8 + q;
      const float adj = hCur[XI(i, col)] * (1.f + rv[q]) / dn;
      hCur[XI(i, col)]  = adj;   // hCur now holds adj
      htLds[XI(i, col)] = adj;   // GS starting point
    }
    __syncthreads();

    // (2) s = v @ (adj - pi) = v@adj - bias   [WMMA]
    {
      v8f acc = wmma_vx(aFrag, hCur, half, lr);
#pragma unroll
      for (int r = 0; r < 8; ++r)
        sLds[XI(mb + r, lr)] = acc[r] - biasReg[r];
    }
    __syncthreads();

    // (3) delta, then judge = any_k all_i |s + v[:,k]*delta_k| < absb+EPS
#pragma unroll
    for (int q = 0; q < 8; ++q) {
      const int i = ig * 8 + q;
      const float s  = sLds[XI(i, col)];
      const float ab = absbLds[i];
      const float vd = vLds[i * VP + i];
      dLds[XI(i, col)] =
          (s > ab) ? (ab - s) / vd : ((s < -ab) ? (-ab - s) / vd : 0.f);
    }
    if (tid < BT) judgeLds[tid] = 0.f;
    __syncthreads();
    {
      bool found = false;
      for (int k = ig * 8; k < ig * 8 + 8 && !found; ++k) {
        const float dk = dLds[XI(k, col)];
        bool pass = true;
        for (int i = 0; i < DD; ++i) {
          const float sn = sLds[XI(i, col)] + vLds[i * VP + k] * dk;
          const float ab = absbLds[i] + EPSF;
          if (!(sn < ab && sn > -ab)) { pass = false; break; }
        }
        found = pass;
      }
      if (found) judgeLds[col] = 1.f;   // benign same-value race
    }
    __syncthreads();

    // (4) Gauss-Seidel sweep with incremental rank-1 s updates
    for (int j = 0; j < DD; ++j) {
      if (tid < 32) {                    // wave 0
        float dj = 0.f;
        if (tid < BT) {
          const float sj  = sLds[XI(j, tid)];
          const float ab  = absbLds[j];
          const float vjj = vLds[j * VP + j];
          dj = (sj > ab) ? (ab - sj) / vjj
                         : ((sj < -ab) ? (-ab - sj) / vjj : 0.f);
          djLds[tid] = dj;
          htLds[XI(j, tid)] += dj;
        }
        const unsigned long long bm = __ballot(dj != 0.f);
        if (tid == 0) flagPar[j & 1] = (int)bm;
      }
      __syncthreads();
      if (flagPar[j & 1]) {              // uniform branch (shared scalar)
        const float dj = djLds[col];
        if (dj != 0.f) {
#pragma unroll
          for (int q = 0; q < 8; ++q) {
            const int i = ig * 8 + q;
            sLds[XI(i, col)] += vLds[i * VP + j] * dj;
          }
        }
        __syncthreads();
      }
    }

    // (5) bisection: h_in = pi_bar, h_out = (1-judge)*adj, 10 halvings
    const float jd = judgeLds[col];
#pragma unroll
    for (int q = 0; q < 8; ++q) {
      const int i = ig * 8 + q;
      hinLds[XI(i, col)]  = targLds[i];
      houtLds[XI(i, col)] = (1.f - jd) * hCur[XI(i, col)];
    }
    if (tid == 0) {
      int aj = 1;
      for (int c = 0; c < BT; ++c) aj &= (judgeLds[c] != 0.f);
      allJ = aj;
    }
    __syncthreads();
    if (!allJ) {                         // uniform skip if all columns judged
      for (int it = 0; it < 10; ++it) {
#pragma unroll
        for (int q = 0; q < 8; ++q) {
          const int i = ig * 8 + q;
          const float hi = hinLds[XI(i, col)];
          hmLds[XI(i, col)] = hi + (houtLds[XI(i, col)] - hi) * 0.5f;
        }
        if (tid < BT) jmaskLds[tid] = ~0;
        __syncthreads();
        v8f acc = wmma_vx(aFrag, hmLds, half, lr);   // v@hm
        {
          bool okb = true;
#pragma unroll
          for (int r = 0; r < 8; ++r) {
            const float sv = acc[r] - biasReg[r];    // sm = v@(hm-pi)
            const float ab = absbReg[r] + EPSF;
            okb = okb && (sv <= ab) && (sv >= -ab);
          }
          atomicAnd(&jmaskLds[lr], okb ? ~0 : 0);
        }
        __syncthreads();
        const int jj = jmaskLds[col];
#pragma unroll
        for (int q = 0; q < 8; ++q) {
          const int i = ig * 8 + q;
          if (jj) hinLds[XI(i, col)]  = hmLds[XI(i, col)];
          else    houtLds[XI(i, col)] = hmLds[XI(i, col)];
        }
        __syncthreads();
      }
    }

    // (6) h_next = judge ? h_temp : h_m ; emit to output[:, t+1, :]
#pragma unroll
    for (int q = 0; q < 8; ++q) {
      const int i = ig * 8 + q;
      const float hn = (judgeLds[col] > 0.5f) ? htLds[XI(i, col)]
                                              : hmLds[XI(i, col)];
      hCur[XI(i, col)] = hn;
      out[i * TB + (t + 1) * BB + gb] = hn;
    }
    __syncthreads();
  }

  // ---- h_last ----
#pragma unroll
  for (int q = 0; q < 8; ++q) {
    const int i = ig * 8 + q;
    hlast[i * BB + gb] = hCur[XI(i, col)];
  }
}

extern "C" void kernel_launch(void* const* d_in, const int* in_sizes, int n_in,
                              void* d_out, int out_size, void* d_ws, size_t ws_size,
                              hipStream_t stream) {
  (void)in_sizes; (void)n_in; (void)out_size; (void)d_ws; (void)ws_size;
  const float* inp    = (const float*)d_in[0];
  const float* target = (const float*)d_in[1];
  const float* rets   = (const float*)d_in[2];
  // d_in[3] = hidden (unused by reference: h0 = input[:,0,:])
  const float* W      = (const float*)d_in[4];
  const float* bvec   = (const float*)d_in[5];
  float* out   = (float*)d_out;
  float* hlast = out + (size_t)DD * TT * BB;
  dim3 grid(BB / BT), block(256);
  hipLaunchKernelGGL(notrade_rnn_kernel, grid, block, 0, stream,
                     inp, target, rets, W, bvec, out, hlast);
}